// Unbiased_31275951849581
// MI455X (gfx1250) — compile-verified
//
#include <hip/hip_runtime.h>
#include <hip/hip_bf16.h>

typedef float v2f __attribute__((ext_vector_type(2)));
typedef float v8f __attribute__((ext_vector_type(8)));

constexpr int GB   = 512;      // streaming blocks
constexpr int BAGS = 128;
constexpr int CLS  = 128;
constexpr int S2   = 129;      // padded LDS stride for the inversion matrices

// ---------------------------------------------------------------------------
// K1: streaming pass. Each wave32 owns whole rows (float4 per lane = 128 cols).
// Accumulate neg_log_p into per-block LDS bag accumulator via ds_add_f32,
// then dump per-block partial sums (no global atomics).
// ---------------------------------------------------------------------------
__global__ void k_stream(const float* __restrict__ yp, const int* __restrict__ yt,
                         float* __restrict__ psums, float* __restrict__ pcnt, int N) {
    extern __shared__ float lds[];           // 16384 acc + 128 cnt
    float* acc = lds;
    float* cnt = lds + BAGS * CLS;
    for (int i = threadIdx.x; i < BAGS * CLS + BAGS; i += 256) lds[i] = 0.0f;
    __syncthreads();

    const int lane = threadIdx.x & 31;
    const int wid  = (blockIdx.x * 256 + threadIdx.x) >> 5;   // global wave id
    const int nw   = (GB * 256) >> 5;                         // total waves
    const float4* yp4 = reinterpret_cast<const float4*>(yp);
    const int nchunk = (N + 7) >> 3;

    for (int gc = wid; gc < nchunk; gc += nw) {
        float4 v[8];
        int    bg[8];
#pragma unroll
        for (int j = 0; j < 8; ++j) {        // 8 independent 512B loads in flight
            int row = gc * 8 + j;
            if (row < N) {
                v[j]  = yp4[(size_t)row * 32 + lane];
                bg[j] = yt[row];
            }
        }
#pragma unroll
        for (int j = 0; j < 8; ++j) {
            int row = gc * 8 + j;
            if (row >= N) continue;
            float m = fmaxf(fmaxf(v[j].x, v[j].y), fmaxf(v[j].z, v[j].w));
            for (int o = 16; o > 0; o >>= 1) m = fmaxf(m, __shfl_xor(m, o));
            float s = __expf(v[j].x - m) + __expf(v[j].y - m) +
                      __expf(v[j].z - m) + __expf(v[j].w - m);
            for (int o = 16; o > 0; o >>= 1) s += __shfl_xor(s, o);
            float lse = m + __logf(s);
            int bag = bg[j];
            float* dst = acc + bag * CLS + lane * 4;
            atomicAdd(dst + 0, lse - v[j].x);   // ds_add_f32 (no return)
            atomicAdd(dst + 1, lse - v[j].y);
            atomicAdd(dst + 2, lse - v[j].z);
            atomicAdd(dst + 3, lse - v[j].w);
            if (lane == 0) atomicAdd(cnt + bag, 1.0f);
        }
    }
    __syncthreads();
    float* myp = psums + (size_t)blockIdx.x * (BAGS * CLS);
    for (int i = threadIdx.x; i < BAGS * CLS; i += 256) myp[i] = acc[i];
    if (threadIdx.x < BAGS) pcnt[blockIdx.x * BAGS + threadIdx.x] = cnt[threadIdx.x];
}

// ---------------------------------------------------------------------------
// K2: invert theta (128x128, well-conditioned) via Newton-Schulz iteration
// using V_WMMA_F32_16X16X4_F32. One block, 8 waves; wave w owns output rows
// [16w,16w+16). All operands live in LDS with stride 129 (bank-conflict-free
// column reads for the A fragments).
// ---------------------------------------------------------------------------
__device__ __forceinline__ void wave_gemm128(const float* __restrict__ M,
                                             const float* __restrict__ Nm,
                                             v8f acc[8], int w, int lr, int hi) {
    for (int kk = 0; kk < 32; ++kk) {
        const int k0 = kk * 4;
        // A-frag (16x4 f32): lanes 0-15 -> K = k0,k0+1 ; lanes 16-31 -> K = k0+2,k0+3
        v2f a;
        a.x = M[(16 * w + lr) * S2 + k0 + 2 * hi];
        a.y = M[(16 * w + lr) * S2 + k0 + 2 * hi + 1];
#pragma unroll
        for (int t = 0; t < 8; ++t) {
            // B-frag (4x16 f32): vgpr0 rows K=k0(+2*hi), vgpr1 rows K=k0+1(+2*hi)
            v2f bb;
            bb.x = Nm[(k0 + 2 * hi) * S2 + 16 * t + lr];
            bb.y = Nm[(k0 + 2 * hi + 1) * S2 + 16 * t + lr];
            acc[t] = __builtin_amdgcn_wmma_f32_16x16x4_f32(
                false, a, false, bb, (short)0, acc[t], false, false);
        }
    }
}

__global__ void k_inv(const float* __restrict__ theta, float* __restrict__ thinv) {
    extern __shared__ float lds[];
    float* A = lds;                 // theta          128 x S2
    float* X = A + 128 * S2;        // iterate        128 x S2
    float* T = X + 128 * S2;        // temp A*X       128 x S2
    const int tid = threadIdx.x;

    for (int i = tid; i < 128 * 128; i += 256)
        A[(i >> 7) * S2 + (i & 127)] = theta[i];
    __syncthreads();
    for (int i = tid; i < 128 * 128; i += 256) {
        int r = i >> 7, c = i & 127;
        X[r * S2 + c] = (r == c ? 2.0f : 0.0f) - A[r * S2 + c];   // X0 = 2I - A
    }
    __syncthreads();

    const int lane = tid & 31;
    const int w    = tid >> 5;      // wave id 0..7 -> owns rows 16w..16w+15
    const int lr   = lane & 15;
    const int hi   = lane >> 4;

    for (int it = 0; it < 8; ++it) {
        // T = A * X
        v8f acc[8];
#pragma unroll
        for (int t = 0; t < 8; ++t) acc[t] = (v8f){0.f,0.f,0.f,0.f,0.f,0.f,0.f,0.f};
        wave_gemm128(A, X, acc, w, lr, hi);
#pragma unroll
        for (int t = 0; t < 8; ++t)
#pragma unroll
            for (int r = 0; r < 8; ++r)
                T[(16 * w + r + 8 * hi) * S2 + 16 * t + lr] = acc[t][r];
        __syncthreads();

        // Z = X * T (rows 16w only, reads only own X rows + all of T), X := 2X - Z
#pragma unroll
        for (int t = 0; t < 8; ++t) acc[t] = (v8f){0.f,0.f,0.f,0.f,0.f,0.f,0.f,0.f};
        wave_gemm128(X, T, acc, w, lr, hi);
#pragma unroll
        for (int t = 0; t < 8; ++t)
#pragma unroll
            for (int r = 0; r < 8; ++r) {
                int idx = (16 * w + r + 8 * hi) * S2 + 16 * t + lr;
                X[idx] = 2.0f * X[idx] - acc[t][r];
            }
        __syncthreads();
    }

    for (int i = tid; i < 128 * 128; i += 256)
        thinv[i] = X[(i >> 7) * S2 + (i & 127)];   // row-major inverse: [class][bag]
}

// ---------------------------------------------------------------------------
// K3: one block per bag: reduce partials, scale by 1/count, contract with
// theta_inv column, apply empty-bag -> 1.0.
// ---------------------------------------------------------------------------
__global__ void k_reduce(const float* __restrict__ psums, const float* __restrict__ pcnt,
                         const float* __restrict__ thinv, float* __restrict__ risks) {
    __shared__ float sh[256];
    const int b = blockIdx.x;
    const int t = threadIdx.x;

    float c = 0.0f;
    for (int g = t; g < GB; g += 256) c += pcnt[g * BAGS + b];
    sh[t] = c; __syncthreads();
    for (int o = 128; o > 0; o >>= 1) { if (t < o) sh[t] += sh[t + o]; __syncthreads(); }
    const float count = sh[0];
    __syncthreads();

    float term = 0.0f;
    if (t < CLS) {
        float s0 = 0.f, s1 = 0.f, s2 = 0.f, s3 = 0.f;
        for (int g = 0; g + 3 < GB; g += 4) {
            s0 += psums[(size_t)(g + 0) * (BAGS * CLS) + b * CLS + t];
            s1 += psums[(size_t)(g + 1) * (BAGS * CLS) + b * CLS + t];
            s2 += psums[(size_t)(g + 2) * (BAGS * CLS) + b * CLS + t];
            s3 += psums[(size_t)(g + 3) * (BAGS * CLS) + b * CLS + t];
        }
        float s = (s0 + s1) + (s2 + s3);
        term = (s / fmaxf(count, 1.0f)) * thinv[t * BAGS + b];  // theta_T[c,b]
    }
    sh[t] = term; __syncthreads();
    for (int o = 128; o > 0; o >>= 1) { if (t < o) sh[t] += sh[t + o]; __syncthreads(); }
    if (t == 0) risks[b] = (count > 0.0f) ? sh[0] : 1.0f;
}

__global__ void k_final(const float* __restrict__ risks, float* __restrict__ out) {
    __shared__ float sh[128];
    const int t = threadIdx.x;
    sh[t] = risks[t]; __syncthreads();
    for (int o = 64; o > 0; o >>= 1) { if (t < o) sh[t] += sh[t + o]; __syncthreads(); }
    if (t == 0) out[0] = sh[0];
}

// ---------------------------------------------------------------------------
extern "C" void kernel_launch(void* const* d_in, const int* in_sizes, int n_in,
                              void* d_out, int out_size, void* d_ws, size_t ws_size,
                              hipStream_t stream) {
    const int*   y_true = (const int*)d_in[0];
    const float* y_pred = (const float*)d_in[1];
    const float* theta  = (const float*)d_in[2];
    float*       out    = (float*)d_out;
    const int    N      = in_sizes[0];

    float* ws    = (float*)d_ws;
    float* psums = ws;                               // GB * 16384
    float* pcnt  = psums + (size_t)GB * BAGS * CLS;  // GB * 128
    float* thinv = pcnt + (size_t)GB * BAGS;         // 16384
    float* risks = thinv + BAGS * CLS;               // 128

    const int invLds = 3 * 128 * S2 * (int)sizeof(float);   // ~194 KB (<= 320 KB WGP LDS)
    (void)hipFuncSetAttribute(reinterpret_cast<const void*>(k_inv),
                              hipFuncAttributeMaxDynamicSharedMemorySize, invLds);

    k_stream<<<GB, 256, (BAGS * CLS + BAGS) * sizeof(float), stream>>>(
        y_pred, y_true, psums, pcnt, N);
    k_inv<<<1, 256, invLds, stream>>>(theta, thinv);
    k_reduce<<<BAGS, 256, 0, stream>>>(psums, pcnt, thinv, risks);
    k_final<<<1, 128, 0, stream>>>(risks, out);
}